// ABAE_56367150793047
// MI455X (gfx1250) — compile-verified
//
#include <hip/hip_runtime.h>
#include <hip/hip_bf16.h>
#include <math.h>

typedef float v2f __attribute__((ext_vector_type(2)));
typedef float v4f __attribute__((ext_vector_type(4)));
typedef float v8f __attribute__((ext_vector_type(8)));

#define B_   512
#define L_   100
#define M_   10
#define D_   256
#define NA_  14

// ---------------------------------------------------------------------------
// Kernel 1: y_s[b,:] = mean_l E_w[pos[b,l], :]
// 256 threads = 4 row-groups x 64 lanes; each lane loads float4 (b128).
// Per row: 2 wave-level global_load_b128 instead of 8 b32 -> 4x fewer VMEM ops.
// ---------------------------------------------------------------------------
__global__ __launch_bounds__(256) void ys_kernel(const int* __restrict__ pos,
                                                 const float* __restrict__ E_w,
                                                 float* __restrict__ y_s) {
    const int b = blockIdx.x, t = threadIdx.x;
    const int g = t >> 6;          // row-group 0..3
    const int tg = t & 63;         // covers d components [tg*4, tg*4+4)
    const int* idx = pos + b * L_;

    v4f acc = {0.f, 0.f, 0.f, 0.f};
    #pragma unroll 5
    for (int l = g; l < L_; l += 4) {
        const v4f r = *(const v4f*)(E_w + (size_t)idx[l] * D_ + tg * 4);
        acc += r;
    }

    __shared__ v4f sm[256];
    sm[t] = acc;
    __syncthreads();
    if (g == 0) {
        v4f o = sm[t] + sm[t + 64] + sm[t + 128] + sm[t + 192];
        o *= (1.0f / (float)L_);
        *(v4f*)(y_s + b * D_ + tg * 4) = o;
    }
}

// ---------------------------------------------------------------------------
// Kernel 2: My = y_s @ M_w^T + M_b  via V_WMMA_F32_16X16X4_F32 (fp32-exact).
// One wave per 16x16 output tile; 64 K-steps of K=4. (Verified codegen: 8x
// v_wmma_f32_16x16x4_f32 per unrolled iteration with b64 fragment loads.)
// ---------------------------------------------------------------------------
__global__ __launch_bounds__(32) void wmma_My(const float* __restrict__ y_s,
                                              const float* __restrict__ M_w,
                                              const float* __restrict__ M_b,
                                              float* __restrict__ My) {
    const int tile = blockIdx.x;        // 32 row-tiles x 16 col-tiles
    const int bt   = tile >> 4;
    const int nt   = tile & 15;
    const int lane = threadIdx.x;
    const int half = lane >> 4;
    const int lm   = lane & 15;
    const int m    = bt * 16 + lm;      // A row for this lane
    const int n    = nt * 16 + lm;      // B col for this lane

    v8f c = {};
    for (int k0 = 0; k0 < D_; k0 += 4) {
        const int k = k0 + 2 * half;
        v2f a; a.x = y_s[m * D_ + k];  a.y = y_s[m * D_ + k + 1];
        v2f b; b.x = M_w[n * D_ + k];  b.y = M_w[n * D_ + k + 1];  // B[k][n] = M_w[n][k]
        c = __builtin_amdgcn_wmma_f32_16x16x4_f32(
                /*neg_a=*/false, a, /*neg_b=*/false, b,
                /*c_mod=*/(short)0, c, /*reuse_a=*/false, /*reuse_b=*/false);
    }

    const int col  = nt * 16 + lm;
    const int row0 = bt * 16 + half * 8;
    const float bias = M_b[col];
    #pragma unroll
    for (int r = 0; r < 8; ++r)
        My[(row0 + r) * D_ + col] = c[r] + bias;
}

// ---------------------------------------------------------------------------
// Kernel 3: w[b,l] = exp(tanh(dot(E_w[pos[b,l]], My[b]))); atomic global sum.
// Block per b (8 waves); wave per row; 2x b128 loads + float4 LDS dot.
// ---------------------------------------------------------------------------
__global__ __launch_bounds__(256) void di_kernel(const int* __restrict__ pos,
                                                 const float* __restrict__ E_w,
                                                 const float* __restrict__ My,
                                                 float* __restrict__ wbuf,
                                                 float* __restrict__ gsum) {
    const int b = blockIdx.x, t = threadIdx.x;
    const int lane = t & 31, wave = t >> 5;
    __shared__ v4f Mys4[64];
    if (t < 64) Mys4[t] = *(const v4f*)(My + b * D_ + t * 4);
    __syncthreads();

    const v4f m0 = Mys4[lane];
    const v4f m1 = Mys4[lane + 32];

    float wsum = 0.f;
    for (int l = wave; l < L_; l += 8) {
        const int v = pos[b * L_ + l];
        const float* row = E_w + (size_t)v * D_;
        const v4f r0 = *(const v4f*)(row + lane * 4);
        const v4f r1 = *(const v4f*)(row + 128 + lane * 4);
        float s = r0.x * m0.x + r0.y * m0.y + r0.z * m0.z + r0.w * m0.w
                + r1.x * m1.x + r1.y * m1.y + r1.z * m1.z + r1.w * m1.w;
        #pragma unroll
        for (int off = 16; off > 0; off >>= 1)
            s += __shfl_down(s, off, 32);
        if (lane == 0) {
            const float w = expf(tanhf(s));
            wbuf[b * L_ + l] = w;
            wsum += w;
        }
    }
    if (lane == 0) atomicAdd(gsum, wsum);
}

// ---------------------------------------------------------------------------
// Kernel 4: z_s = l2norm(sum_l a[b,l] * e[b,l,:]); p_t = softmax(z_s@lin_w^T+b);
//           r_s = l2norm(p_t @ T_w).  Fully fused, block per b.
// Gather phase uses the same 4-group x 64-lane float4 structure.
// ---------------------------------------------------------------------------
__global__ __launch_bounds__(256) void zs_kernel(const int* __restrict__ pos,
                                                 const float* __restrict__ E_w,
                                                 const float* __restrict__ wbuf,
                                                 const float* __restrict__ gsum,
                                                 const float* __restrict__ lin_w,
                                                 const float* __restrict__ lin_b,
                                                 const float* __restrict__ T_w,
                                                 float* __restrict__ r_s,
                                                 float* __restrict__ z_s) {
    const int b = blockIdx.x, t = threadIdx.x;
    const int lane = t & 31, wave = t >> 5;
    const int g = t >> 6, tg = t & 63;

    __shared__ float sw[L_];
    __shared__ float red[8];
    __shared__ float zsh[D_];
    __shared__ float pt[16];
    __shared__ v4f sm[256];

    const float inv = 1.0f / gsum[0];
    if (t < L_) sw[t] = wbuf[b * L_ + t] * inv;
    __syncthreads();

    const int* idx = pos + b * L_;
    v4f acc = {0.f, 0.f, 0.f, 0.f};
    #pragma unroll 5
    for (int l = g; l < L_; l += 4) {
        const v4f r = *(const v4f*)(E_w + (size_t)idx[l] * D_ + tg * 4);
        acc += r * sw[l];
    }
    sm[t] = acc;
    __syncthreads();

    v4f o = {0.f, 0.f, 0.f, 0.f};
    float sq = 0.f;
    if (g == 0) {
        o = sm[t] + sm[t + 64] + sm[t + 128] + sm[t + 192];
        sq = o.x * o.x + o.y * o.y + o.z * o.z + o.w * o.w;
    }
    #pragma unroll
    for (int off = 16; off > 0; off >>= 1) sq += __shfl_down(sq, off, 32);
    if (lane == 0) red[wave] = sq;
    __syncthreads();
    if (t == 0) {
        float s = 0.f;
        for (int i = 0; i < 8; ++i) s += red[i];
        red[0] = fmaxf(sqrtf(s), 1e-12f);
    }
    __syncthreads();
    if (g == 0) {
        const float invn = 1.0f / red[0];
        const v4f z = o * invn;
        zsh[tg * 4 + 0] = z.x; zsh[tg * 4 + 1] = z.y;
        zsh[tg * 4 + 2] = z.z; zsh[tg * 4 + 3] = z.w;
        *(v4f*)(z_s + b * D_ + tg * 4) = z;
    }
    __syncthreads();

    // logits (14) from LDS
    if (t < NA_) {
        float lg = lin_b[t];
        for (int j = 0; j < D_; ++j) lg += zsh[j] * lin_w[t * D_ + j];
        pt[t] = lg;
    }
    __syncthreads();
    if (t == 0) {
        float mx = pt[0];
        for (int j = 1; j < NA_; ++j) mx = fmaxf(mx, pt[j]);
        float s = 0.f;
        for (int j = 0; j < NA_; ++j) { const float e = expf(pt[j] - mx); pt[j] = e; s += e; }
        const float is = 1.0f / s;
        for (int j = 0; j < NA_; ++j) pt[j] *= is;
    }
    __syncthreads();

    // r_s = l2norm(p_t @ T_w)  (one component per thread)
    float r = 0.f;
    #pragma unroll
    for (int j = 0; j < NA_; ++j) r += pt[j] * T_w[j * D_ + t];
    float rq = r * r;
    #pragma unroll
    for (int off = 16; off > 0; off >>= 1) rq += __shfl_down(rq, off, 32);
    if (lane == 0) red[wave] = rq;
    __syncthreads();
    if (t == 0) {
        float s = 0.f;
        for (int i = 0; i < 8; ++i) s += red[i];
        red[0] = fmaxf(sqrtf(s), 1e-12f);
    }
    __syncthreads();
    r_s[b * D_ + t] = r / red[0];
}

// ---------------------------------------------------------------------------
// Kernel 5: z_n[b,m,:] = l2norm(mean_l E_w[negs[b,m,l], :])  — dominant gather.
// 5120 blocks; 4 row-groups x 64 lanes x float4 (b128). E_w is L2-resident.
// ---------------------------------------------------------------------------
__global__ __launch_bounds__(256) void zn_kernel(const int* __restrict__ negs,
                                                 const float* __restrict__ E_w,
                                                 float* __restrict__ z_n) {
    const int bm = blockIdx.x;          // b*M + m
    const int t  = threadIdx.x;
    const int lane = t & 31, wave = t >> 5;
    const int g = t >> 6, tg = t & 63;

    __shared__ v4f sm[256];
    __shared__ float red[8];
    const int* idx = negs + (size_t)bm * L_;

    v4f acc = {0.f, 0.f, 0.f, 0.f};
    #pragma unroll 5
    for (int l = g; l < L_; l += 4) {
        const v4f r = *(const v4f*)(E_w + (size_t)idx[l] * D_ + tg * 4);
        acc += r;
    }
    sm[t] = acc;
    __syncthreads();

    v4f o = {0.f, 0.f, 0.f, 0.f};
    float sq = 0.f;
    if (g == 0) {
        o = sm[t] + sm[t + 64] + sm[t + 128] + sm[t + 192];
        o *= (1.0f / (float)L_);
        sq = o.x * o.x + o.y * o.y + o.z * o.z + o.w * o.w;
    }
    #pragma unroll
    for (int off = 16; off > 0; off >>= 1) sq += __shfl_down(sq, off, 32);
    if (lane == 0) red[wave] = sq;
    __syncthreads();
    if (t == 0) {
        float s = 0.f;
        for (int i = 0; i < 8; ++i) s += red[i];
        red[0] = fmaxf(sqrtf(s), 1e-12f);
    }
    __syncthreads();
    if (g == 0) {
        const float invn = 1.0f / red[0];
        *(v4f*)(z_n + (size_t)bm * D_ + tg * 4) = o * invn;
    }
}

// ---------------------------------------------------------------------------
extern "C" void kernel_launch(void* const* d_in, const int* in_sizes, int n_in,
                              void* d_out, int out_size, void* d_ws, size_t ws_size,
                              hipStream_t stream) {
    const int*   pos   = (const int*)  d_in[0];
    const int*   negs  = (const int*)  d_in[1];
    const float* E_w   = (const float*)d_in[2];
    const float* T_w   = (const float*)d_in[3];
    const float* M_w   = (const float*)d_in[4];
    const float* M_b   = (const float*)d_in[5];
    const float* lin_w = (const float*)d_in[6];
    const float* lin_b = (const float*)d_in[7];

    float* out = (float*)d_out;
    float* r_s = out;                         // [B, d]
    float* z_s = out + B_ * D_;               // [B, d]
    float* z_n = out + 2 * B_ * D_;           // [B, M, d]

    char* ws = (char*)d_ws;
    float* y_s  = (float*)ws; ws += (size_t)B_ * D_ * sizeof(float);
    float* My   = (float*)ws; ws += (size_t)B_ * D_ * sizeof(float);
    float* wbuf = (float*)ws; ws += (size_t)B_ * L_ * sizeof(float);
    float* gsum = (float*)ws; ws += 256;

    hipMemsetAsync(gsum, 0, sizeof(float), stream);

    ys_kernel<<<B_, 256, 0, stream>>>(pos, E_w, y_s);
    wmma_My<<<(B_ / 16) * (D_ / 16), 32, 0, stream>>>(y_s, M_w, M_b, My);
    di_kernel<<<B_, 256, 0, stream>>>(pos, E_w, My, wbuf, gsum);
    zs_kernel<<<B_, 256, 0, stream>>>(pos, E_w, wbuf, gsum, lin_w, lin_b, T_w, r_s, z_s);
    zn_kernel<<<B_ * M_, 256, 0, stream>>>(negs, E_w, z_n);
}